// Repro_79422535238141
// MI455X (gfx1250) — compile-verified
//
#include <hip/hip_runtime.h>

// CDNA5 / gfx1250 banded-GEMM antialiased bilinear resize, full-TDM edition.
// Pass 1 (horizontal): X(2070x456) -> mid(2070x272); per tile:
//   TDM-load 16x32 f32 (row stride 456) into LDS (padded 34-dword rows),
//   scatter the 5-tap weight band into LDS, 8 chained V_WMMA_F32_16X16X4_F32,
//   stage D(16x16) in LDS, TDM-store to mid (OOB rows dropped by the TDM).
// Pass 2 (vertical): per plane, Wv(16x32) @ mid-tile(32x16) -> out(271x272),
//   same structure with the weight band as the A operand.
// TDM OOB reads return zero (matches zero resize weights outside the window)
// and OOB writes are dropped (replaces ragged-edge store masking).

typedef __attribute__((ext_vector_type(2))) float v2f;
typedef __attribute__((ext_vector_type(8))) float v8f;
typedef __attribute__((ext_vector_type(4))) unsigned int u32x4;
typedef __attribute__((ext_vector_type(8))) unsigned int u32x8;

namespace {
constexpr int IN_H  = 345;
constexpr int IN_W  = 456;
constexpr int MID_W = 272;   // width after horizontal pass
constexpr int OUT_H = 271;
constexpr int OUT_W = 272;
constexpr int NP    = 6;               // 2*3 planes
constexpr int R1    = NP * IN_H;       // 2070 flattened rows for horizontal pass
constexpr int KS    = 5;               // max taps
constexpr int BANDP = 34;              // padded band stride in dwords
}

// Build the two D# SGPR groups for a 2-D tile (tile_d0 x tile_d1 elements of
// 4 bytes, source/dest row stride stride0 elements). If pad==true, the TDM
// inserts 2 dwords of LDS padding after every 32 dwords (loads only).
__device__ __forceinline__ void tdm_desc_2d(u32x4& g0, u32x8& g1,
                                            unsigned lds_addr, const void* gaddr,
                                            unsigned tensor_d0, unsigned tensor_d1,
                                            unsigned tile_d0, unsigned tile_d1,
                                            unsigned stride0, bool pad)
{
    unsigned long long ga = (unsigned long long)(size_t)gaddr;
    g0.x = 1u;                                            // count=1 (valid user D#)
    g0.y = lds_addr;                                      // LDS byte address
    g0.z = (unsigned)(ga & 0xFFFFFFFFull);                // global_addr[31:0]
    g0.w = (unsigned)((ga >> 32) & 0x1FFFFFFull) | (2u << 30); // addr[56:32] | type=2
    g1[0] = (2u << 16)                                    // data_size = 4 bytes
          | (pad ? ((1u << 20)                            // pad_enable
                  | (4u << 22)                            // pad_interval: 32 dwords
                  | (1u << 25)) : 0u);                    // pad_amount: 2 dwords
    g1[1] = (tensor_d0 & 0xFFFFu) << 16;                  // tensor_dim0[15:0]
    g1[2] = ((tensor_d0 >> 16) & 0xFFFFu) | ((tensor_d1 & 0xFFFFu) << 16);
    g1[3] = ((tensor_d1 >> 16) & 0xFFFFu) | (tile_d0 << 16); // tile_dim0
    g1[4] = tile_d1 & 0xFFFFu;                            // tile_dim1 (tile_dim2 = 0)
    g1[5] = stride0;                                      // tensor_dim0_stride[31:0]
    g1[6] = 0u;
    g1[7] = 0u;
}

__device__ __forceinline__ void tdm_load_2d(unsigned lds_addr, const void* gaddr,
                                            unsigned tensor_d0, unsigned tensor_d1,
                                            unsigned tile_d0, unsigned tile_d1,
                                            unsigned stride0, bool pad)
{
    u32x4 g0; u32x8 g1;
    tdm_desc_2d(g0, g1, lds_addr, gaddr, tensor_d0, tensor_d1,
                tile_d0, tile_d1, stride0, pad);
    asm volatile("tensor_load_to_lds %0, %1" :: "s"(g0), "s"(g1) : "memory");
}

__device__ __forceinline__ void tdm_store_2d(unsigned lds_addr, void* gaddr,
                                             unsigned tensor_d0, unsigned tensor_d1,
                                             unsigned tile_d0, unsigned tile_d1,
                                             unsigned stride0)
{
    u32x4 g0; u32x8 g1;
    tdm_desc_2d(g0, g1, lds_addr, gaddr, tensor_d0, tensor_d1,
                tile_d0, tile_d1, stride0, false);
    asm volatile("tensor_store_from_lds %0, %1" :: "s"(g0), "s"(g1) : "memory");
}

// Reference-exact tap computation (float32 math, truncating int casts,
// clamp_min/clamp_max, tap masking by window size, weight normalization).
__device__ __forceinline__ void calc_taps(int o, float scale, float invscale, int in_size,
                                          int &xmin, float &w0, float &w1, float &w2,
                                          float &w3, float &w4)
{
    float center = ((float)o + 0.5f) * scale;
    int lo = (int)(center - scale + 0.5f);   // trunc toward zero == astype(int64)
    if (lo < 0) lo = 0;
    int hi = (int)(center + scale + 0.5f);
    if (hi > in_size) hi = in_size;
    int size = hi - lo;
    if (size > KS) size = KS;
    xmin = lo;

    float w[KS];
    float s = 0.0f;
#pragma unroll
    for (int j = 0; j < KS; ++j) {
        float t = fabsf(((float)(j + lo) - center + 0.5f) * invscale);
        t = 1.0f - fminf(t, 1.0f);
        t = (j < size) ? t : 0.0f;           // mask taps past window
        w[j] = t;
        s += t;
    }
    float inv = 1.0f / s;
    w0 = w[0] * inv; w1 = w[1] * inv; w2 = w[2] * inv; w3 = w[3] * inv; w4 = w[4] * inv;
}

// Zero band columns 0..31 of all 16 rows, then scatter this lane's 5 taps at
// [idx][off..off+4]. DS ops from one wave execute in order; __syncthreads()
// provides the compiler-level ordering across lanes (lowers to a wave barrier).
__device__ __forceinline__ void stage_band(float* wband, int lane, int idx, int off,
                                           float w0, float w1, float w2, float w3, float w4)
{
    float* z = &wband[(lane >> 1) * BANDP + (lane & 1) * 16];
#pragma unroll
    for (int j = 0; j < 16; ++j) z[j] = 0.0f;
    __syncthreads();
    float* p = &wband[idx * BANDP + off];
    p[0] = w0; p[1] = w1; p[2] = w2; p[3] = w3; p[4] = w4;
    __syncthreads();
}

// ---------------- Pass 1: horizontal 456 -> 272 ----------------
__global__ void __launch_bounds__(32) resize_h_wmma(const float* __restrict__ x,
                                                    float* __restrict__ mid)
{
    __shared__ float tile[16 * BANDP];    // X tile in, reused as D tile out
    __shared__ float wband[16 * BANDP];   // weight band, [out-col][band pos]

    const int lane = (int)threadIdx.x;
    const int hf   = lane >> 4;           // 0: lanes 0-15, 1: lanes 16-31
    const int lm   = lane & 15;
    const int r0   = (int)blockIdx.x * 16;   // flattened image-row tile
    const int o0   = (int)blockIdx.y * 16;   // output-column tile

    const float scale    = (float)IN_W / (float)MID_W;   // 1.6764706
    const float invscale = (float)MID_W / (float)IN_W;

    // Tile band base = xmin of the first output column (xmin monotonic in o).
    float c0 = ((float)o0 + 0.5f) * scale;
    int base = (int)(c0 - scale + 0.5f);
    if (base < 0) base = 0;

    // Async TDM: 16 rows x 32 cols from (r0, base), row stride 456.
    // tensor dims = remaining extent so OOB rows/cols zero-fill at the edges.
    tdm_load_2d((unsigned)(size_t)tile, x + (size_t)r0 * IN_W + base,
                (unsigned)(IN_W - base), (unsigned)(R1 - r0),
                32u, 16u, (unsigned)IN_W, true);

    int xmin; float w0, w1, w2, w3, w4;
    calc_taps(o0 + lm, scale, invscale, IN_W, xmin, w0, w1, w2, w3, w4);
    stage_band(wband, lane, lm, xmin - base, w0, w1, w2, w3, w4);

    __builtin_amdgcn_s_wait_tensorcnt(0);
    asm volatile("" ::: "memory");

    v8f acc = {};
#pragma unroll
    for (int kk = 0; kk < 8; ++kk) {
        const int k0 = 4 * kk + 2 * hf;       // this lane's K pair in the 32-band
        v2f A, B;
        A.x = tile[lm * BANDP + k0];          // A[m, k] = x[r0+m, base+k]
        A.y = tile[lm * BANDP + k0 + 1];
        B.x = wband[lm * BANDP + k0];         // B[k, n] = W[base+k, o0+n]
        B.y = wband[lm * BANDP + k0 + 1];
        acc = __builtin_amdgcn_wmma_f32_16x16x4_f32(false, A, false, B,
                                                    (short)0, acc, false, false);
    }

    // Stage D(16x16) in LDS (reusing the input tile) and TDM-store it.
    float* dtile = tile;                      // same-wave DS ordering keeps this safe
#pragma unroll
    for (int i = 0; i < 8; ++i)
        dtile[(i + 8 * hf) * 16 + lm] = acc[i];   // C/D: vgpr i -> row i (+8 hi half)
    asm volatile("s_wait_dscnt 0" ::: "memory");  // TDM reads LDS; stores must be done
    tdm_store_2d((unsigned)(size_t)dtile, mid + (size_t)r0 * MID_W + o0,
                 (unsigned)(MID_W - o0), (unsigned)(R1 - r0),   // OOB rows dropped
                 16u, 16u, (unsigned)MID_W);
    // S_ENDPGM's implicit S_WAIT_IDLE covers TENSORcnt.
}

// ---------------- Pass 2: vertical 345 -> 271 ----------------
__global__ void __launch_bounds__(32) resize_v_wmma(const float* __restrict__ mid,
                                                    float* __restrict__ out)
{
    __shared__ float tile[32 * 16];       // mid tile in (32 rows x 16 cols), D tile out
    __shared__ float wband[16 * BANDP];   // weight band, [out-row][band pos]

    const int lane = (int)threadIdx.x;
    const int hf   = lane >> 4;
    const int lm   = lane & 15;
    const int co0  = (int)blockIdx.x * 16;   // column tile (272/16 exact)
    const int ro0  = (int)blockIdx.y * 16;   // output-row tile
    const int p    = (int)blockIdx.z;        // plane (n*c)

    const float scale    = (float)IN_H / (float)OUT_H;   // 1.2730627
    const float invscale = (float)OUT_H / (float)IN_H;

    float c0 = ((float)ro0 + 0.5f) * scale;
    int base = (int)(c0 - scale + 0.5f);
    if (base < 0) base = 0;

    // Async TDM: 32 rows x 16 cols from (base, co0) of this plane, stride 272.
    tdm_load_2d((unsigned)(size_t)tile,
                mid + ((size_t)p * IN_H + base) * MID_W + co0,
                (unsigned)(MID_W - co0), (unsigned)(IN_H - base),
                16u, 32u, (unsigned)MID_W, false);

    int xmin; float w0, w1, w2, w3, w4;
    calc_taps(ro0 + lm, scale, invscale, IN_H, xmin, w0, w1, w2, w3, w4);
    stage_band(wband, lane, lm, xmin - base, w0, w1, w2, w3, w4);

    __builtin_amdgcn_s_wait_tensorcnt(0);
    asm volatile("" ::: "memory");

    v8f acc = {};
#pragma unroll
    for (int kk = 0; kk < 8; ++kk) {
        const int k0 = 4 * kk + 2 * hf;
        v2f A, B;
        A.x = wband[lm * BANDP + k0];         // A[m, k] = Wv[ro0+m, base+k]
        A.y = wband[lm * BANDP + k0 + 1];
        B.x = tile[k0 * 16 + lm];             // B[k, n] = mid[base+k, co0+n]
        B.y = tile[(k0 + 1) * 16 + lm];
        acc = __builtin_amdgcn_wmma_f32_16x16x4_f32(false, A, false, B,
                                                    (short)0, acc, false, false);
    }

    // Stage D(16x16) in LDS (reusing the tile buffer) and TDM-store to out;
    // the ragged last row tile (271 % 16) is dropped by tensor_dim1.
    float* dtile = tile;
#pragma unroll
    for (int i = 0; i < 8; ++i)
        dtile[(i + 8 * hf) * 16 + lm] = acc[i];
    asm volatile("s_wait_dscnt 0" ::: "memory");
    tdm_store_2d((unsigned)(size_t)dtile,
                 out + ((size_t)p * OUT_H + ro0) * OUT_W + co0,
                 (unsigned)(OUT_W - co0), (unsigned)(OUT_H - ro0),
                 16u, 16u, (unsigned)OUT_W);
}

extern "C" void kernel_launch(void* const* d_in, const int* in_sizes, int n_in,
                              void* d_out, int out_size, void* d_ws, size_t ws_size,
                              hipStream_t stream)
{
    const float* x   = (const float*)d_in[0];     // (2,3,345,456) f32
    float*       out = (float*)d_out;             // (2,3,271,272) f32
    float*       mid = (float*)d_ws;              // (2,3,345,272) f32 = 2.25 MB scratch

    dim3 g1((R1 + 15) / 16, MID_W / 16);          // 130 x 17 tiles, 1 wave each
    resize_h_wmma<<<g1, 32, 0, stream>>>(x, mid);

    dim3 g2(OUT_W / 16, (OUT_H + 15) / 16, NP);   // 17 x 17 x 6 tiles
    resize_v_wmma<<<g2, 32, 0, stream>>>(mid, out);
}